// SAGE_63324997812873
// MI455X (gfx1250) — compile-verified
//
#include <hip/hip_runtime.h>
#include <hip/hip_bf16.h>
#include <math.h>

typedef float v2f __attribute__((ext_vector_type(2)));
typedef float v8f __attribute__((ext_vector_type(8)));

// ---------------------------------------------------------------- utilities
__global__ void zero_f32_kernel(float* __restrict__ p, long n) {
    long i = (long)blockIdx.x * blockDim.x + threadIdx.x;
    if (i < n) p[i] = 0.0f;
}

__global__ void degree_kernel(const int* __restrict__ row, float* __restrict__ deg, int E) {
    int e = blockIdx.x * blockDim.x + threadIdx.x;
    if (e < E) atomicAdd(&deg[row[e]], 1.0f);
}

__global__ void rdeg_kernel(float* __restrict__ deg, int n) {
    int i = blockIdx.x * blockDim.x + threadIdx.x;
    if (i < n) deg[i] = 1.0f / fmaxf(deg[i], 1.0f);
}

// Edge-parallel scatter-add: one thread per (edge, 4-feature chunk).
// x rows are L2-resident (<=173MB feature matrix vs 192MB L2), scatter uses
// global_atomic_add_f32 at the L2 atomic units.
__global__ void scatter_add_kernel(const float* __restrict__ x,
                                   const int* __restrict__ row,
                                   const int* __restrict__ col,
                                   float* __restrict__ out,
                                   int total, int lgChunks, int d) {
    int tid = blockIdx.x * blockDim.x + threadIdx.x;
    if (tid >= total) return;
    int e = tid >> lgChunks;
    int c = (tid - (e << lgChunks)) << 2;     // feature offset (x4 floats)
    int src = col[e];
    int dst = row[e];
    const float4 v = *reinterpret_cast<const float4*>(x + (size_t)src * d + c);
    float* o = out + (size_t)dst * d + c;
    atomicAdd(o + 0, v.x);
    atomicAdd(o + 1, v.y);
    atomicAdd(o + 2, v.z);
    atomicAdd(o + 3, v.w);
}

// ---------------------------------------------------------------- fused SAGE GEMM
// out[M,N] = (rdeg[m] * agg[m,:]) @ W + x @ R + bias   via V_WMMA_F32_16X16X4_F32.
// Block = 128 threads = 4 waves; each wave owns one 16x16 output tile along N.
//
// f32 WMMA operand layout (wave32):
//   A (16x4): lane r<16 -> row r, VGPR0=K0 VGPR1=K1; lanes 16-31 -> K2/K3.
//   B (4x16): VGPR0 = K{0|2} row over lanes, VGPR1 = K{1|3}.
//   C/D (16x16 f32, 8 VGPRs): row = v + 8*(lane>>4), col = lane&15.
// OOB tile rows/cols only affect their own D row/col, so loads are clamped
// and only stores are guarded (EXEC stays all-1s around the WMMA).
__global__ void sage_gemm_kernel(const float* __restrict__ agg,
                                 const float* __restrict__ rdeg,
                                 const float* __restrict__ x,
                                 const float* __restrict__ W,
                                 const float* __restrict__ R,
                                 const float* __restrict__ bias,
                                 float* __restrict__ out,
                                 int M, int K, int N) {
    const int lane = threadIdx.x & 31;
    const int wave = threadIdx.x >> 5;
    const int half = lane >> 4;
    const int r    = lane & 15;
    const int m0   = blockIdx.x * 16;
    const int n0   = blockIdx.y * 64 + wave * 16;

    int arow = m0 + r; if (arow >= M) arow = M - 1;     // clamped load row
    int bcol = n0 + r; if (bcol >= N) bcol = N - 1;     // clamped load col

    const float scale = rdeg[arow];                     // fused mean divide
    const float* __restrict__ ap = agg + (size_t)arow * K;
    const float* __restrict__ xp = x   + (size_t)arow * K;

    v8f acc = {0.f, 0.f, 0.f, 0.f, 0.f, 0.f, 0.f, 0.f};

    for (int k0 = 0; k0 < K; k0 += 4) {
        const int kk = k0 + 2 * half;
        v2f a1, a2, b1, b2;
        a1.x = ap[kk]     * scale;
        a1.y = ap[kk + 1] * scale;
        a2.x = xp[kk];
        a2.y = xp[kk + 1];
        b1.x = W[(size_t)kk       * N + bcol];
        b1.y = W[(size_t)(kk + 1) * N + bcol];
        b2.x = R[(size_t)kk       * N + bcol];
        b2.y = R[(size_t)(kk + 1) * N + bcol];
        acc = __builtin_amdgcn_wmma_f32_16x16x4_f32(false, a1, false, b1,
                                                    (short)0, acc, false, false);
        acc = __builtin_amdgcn_wmma_f32_16x16x4_f32(false, a2, false, b2,
                                                    (short)0, acc, false, false);
    }

#pragma unroll
    for (int v = 0; v < 8; ++v) {
        int orow = m0 + v + 8 * half;
        int ocol = n0 + r;
        if (orow < M && ocol < N)
            out[(size_t)orow * N + ocol] = acc[v] + bias[ocol];
    }
}

// ---------------------------------------------------------------- batchnorm
// One feature per lane (blockDim.x == d), coalesced row loads, block-strided
// over nodes, then one atomicAdd per feature per block.
__global__ void bn_stats_kernel(const float* __restrict__ h,
                                float* __restrict__ sums,
                                float* __restrict__ sumsq,
                                int M, int d) {
    int f = threadIdx.x;
    float s = 0.f, q = 0.f;
    for (int i = blockIdx.x; i < M; i += gridDim.x) {
        float v = h[(size_t)i * d + f];
        s += v;
        q += v * v;
    }
    atomicAdd(&sums[f], s);
    atomicAdd(&sumsq[f], q);
}

__global__ void bn_apply_relu_kernel(float* __restrict__ h,
                                     const float* __restrict__ sums,
                                     const float* __restrict__ sumsq,
                                     const float* __restrict__ gamma,
                                     const float* __restrict__ beta,
                                     int M, int d) {
    long i = (long)blockIdx.x * blockDim.x + threadIdx.x;
    if (i >= (long)M * d) return;
    int f = (int)(i & (long)(d - 1));          // d is a power of two (256)
    float invM = 1.0f / (float)M;
    float mu   = sums[f] * invM;
    float var  = fmaxf(sumsq[f] * invM - mu * mu, 0.0f);
    float sc   = gamma[f] * rsqrtf(var + 1e-5f);
    float v    = sc * (h[i] - mu) + beta[f];
    h[i] = fmaxf(v, 0.0f);
}

// ---------------------------------------------------------------- log_softmax
__global__ void log_softmax_kernel(float* __restrict__ out, int M, int C) {
    int i = blockIdx.x * blockDim.x + threadIdx.x;
    if (i >= M) return;
    float* p = out + (size_t)i * C;
    float mx = -3.402823466e38f;
    for (int c = 0; c < C; ++c) mx = fmaxf(mx, p[c]);
    float s = 0.f;
    for (int c = 0; c < C; ++c) s += expf(p[c] - mx);
    float l = logf(s);
    for (int c = 0; c < C; ++c) p[c] = p[c] - mx - l;
}

// ---------------------------------------------------------------- host driver
static inline int ctz_i(int v) { int c = 0; while (!(v & 1)) { v >>= 1; ++c; } return c; }

extern "C" void kernel_launch(void* const* d_in, const int* in_sizes, int n_in,
                              void* d_out, int out_size, void* d_ws, size_t ws_size,
                              hipStream_t stream) {
    const float* x  = (const float*)d_in[0];
    const int*   ei = (const int*)d_in[1];
    const float* W0 = (const float*)d_in[2];
    const float* R0 = (const float*)d_in[3];
    const float* b0 = (const float*)d_in[4];
    const float* g0 = (const float*)d_in[5];
    const float* be0= (const float*)d_in[6];
    const float* W1 = (const float*)d_in[7];
    const float* R1 = (const float*)d_in[8];
    const float* b1 = (const float*)d_in[9];
    const float* g1 = (const float*)d_in[10];
    const float* be1= (const float*)d_in[11];
    const float* W2 = (const float*)d_in[12];
    const float* R2 = (const float*)d_in[13];
    const float* b2 = (const float*)d_in[14];
    float* out = (float*)d_out;

    const int dh  = in_sizes[4];              // 256
    const int din = in_sizes[2] / dh;         // 128
    const int dc  = in_sizes[14];             // 40
    const int M   = in_sizes[0] / din;        // 169343
    const int E   = in_sizes[1] / 2;
    const int* row = ei;
    const int* col = ei + E;

    // -------- workspace carve-up (256B aligned)
    size_t off = 0;
    auto carve = [&](size_t bytes) -> float* {
        float* p = (float*)((char*)d_ws + off);
        off += (bytes + 255) & ~(size_t)255;
        return p;
    };
    float* rdeg  = carve((size_t)M * sizeof(float));
    float* sums  = carve((size_t)dh * sizeof(float));
    float* sumsq = carve((size_t)dh * sizeof(float));
    float* agg   = carve((size_t)M * dh * sizeof(float));   // aggregation buffer
    float* h0    = carve((size_t)M * dh * sizeof(float));
    float* h1    = carve((size_t)M * dh * sizeof(float));
    (void)ws_size; (void)n_in; (void)out_size;

    const int B = 256;
    auto zero = [&](float* p, long n) {
        zero_f32_kernel<<<(unsigned)((n + B - 1) / B), B, 0, stream>>>(p, n);
    };
    auto scatter = [&](const float* src, float* dst, int d) {
        int chunks = d >> 2;
        int total  = E * chunks;
        scatter_add_kernel<<<(total + B - 1) / B, B, 0, stream>>>(
            src, row, col, dst, total, ctz_i(chunks), d);
    };
    auto gemm = [&](const float* a, const float* xx, const float* W,
                    const float* R, const float* bias, float* o, int K, int N) {
        dim3 grid((M + 15) / 16, (N + 63) / 64);
        sage_gemm_kernel<<<grid, 128, 0, stream>>>(a, rdeg, xx, W, R, bias, o, M, K, N);
    };
    auto batchnorm = [&](float* h, const float* g, const float* be) {
        zero(sums, dh); zero(sumsq, dh);
        bn_stats_kernel<<<2048, dh, 0, stream>>>(h, sums, sumsq, M, dh);
        long n = (long)M * dh;
        bn_apply_relu_kernel<<<(unsigned)((n + B - 1) / B), B, 0, stream>>>(
            h, sums, sumsq, g, be, M, dh);
    };

    // -------- degrees (once; graph fixed across layers)
    zero(rdeg, M);
    degree_kernel<<<(E + B - 1) / B, B, 0, stream>>>(row, rdeg, E);
    rdeg_kernel<<<(M + B - 1) / B, B, 0, stream>>>(rdeg, M);

    // -------- layer 0: 128 -> 256, BN + ReLU
    zero(agg, (long)M * din);
    scatter(x, agg, din);
    gemm(agg, x, W0, R0, b0, h0, din, dh);
    batchnorm(h0, g0, be0);

    // -------- layer 1: 256 -> 256, BN + ReLU
    zero(agg, (long)M * dh);
    scatter(h0, agg, dh);
    gemm(agg, h0, W1, R1, b1, h1, dh, dh);
    batchnorm(h1, g1, be1);

    // -------- layer 2: 256 -> 40, log_softmax (writes d_out directly)
    zero(agg, (long)M * dh);
    scatter(h1, agg, dh);
    gemm(agg, h1, W2, R2, b2, out, dh, dc);
    log_softmax_kernel<<<(M + B - 1) / B, B, 0, stream>>>(out, M, dc);
}